// Glm4MoeAttention_75247827026062
// MI455X (gfx1250) — compile-verified
//
#include <hip/hip_runtime.h>
#include <hip/hip_bf16.h>

typedef float v2f __attribute__((ext_vector_type(2)));
typedef float v8f __attribute__((ext_vector_type(8)));

static __device__ __forceinline__ v8f wmma_f32(v2f a, v2f b, v8f c) {
  return __builtin_amdgcn_wmma_f32_16x16x4_f32(
      /*neg_a=*/false, a, /*neg_b=*/false, b,
      /*c_mod=*/(short)0, c, /*reuse_a=*/false, /*reuse_b=*/false);
}

// gfx1250 async direct global->LDS copy (ASYNCcnt path), 16B per lane.
// LDS destination passed as the low 32 bits of the generic shared pointer
#define USE_ASYNC_LDS 1

static __device__ __forceinline__ void async_copy_b128(const float* gsrc,
                                                       float* ldst) {
#if USE_ASYNC_LDS
  unsigned loff = (unsigned)(size_t)ldst;
  asm volatile("global_load_async_to_lds_b128 %0, %1, off" ::"v"(loff),
               "v"(gsrc)
               : "memory");
#else
  *(float4*)ldst = *(const float4*)gsrc;
#endif
}

// Wait until at most N of this wave's async copies remain outstanding.
#if USE_ASYNC_LDS
#define WAIT_ASYNC(n) asm volatile("s_wait_asynccnt %0" ::"i"(n) : "memory")
#else
#define WAIT_ASYNC(n)
#endif

// ---------------------------------------------------------------------------
// GEMM:  C[M,N] = A[M,K] @ W[N,K]^T (+ bias[N])      (f32, WMMA 16x16x4)
// Block: 256 threads = 8 waves, tile 64(M) x 128(N), BK = 32.
// Wave w -> 32x32 sub-tile (wr = w>>2, wc = w&3), 2x2 accumulators:
// per K-step 2 A + 2 B LDS loads feed 4 independent WMMA chains.
// Global->LDS staging is async (ASYNCcnt) and DOUBLE-BUFFERED: chunk i+1
// streams into the other buffer while WMMA consumes chunk i. Each wave
// issues 6 async b128 ops per chunk; async loads complete in order, so
// s_wait_asynccnt<=6 after issuing chunk i+1 guarantees chunk i landed.
// ---------------------------------------------------------------------------
#define GEMM_LDA 36  // padded LDS row stride (floats): conflict-free

__global__ __launch_bounds__(256) void gemm_nt(
    const float* __restrict__ A, const float* __restrict__ W,
    const float* __restrict__ bias, float* __restrict__ C,
    int M, int N, int K) {
  __shared__ float As[2][64 * GEMM_LDA];
  __shared__ float Bs[2][128 * GEMM_LDA];
  const int tid  = threadIdx.x;
  const int wave = tid >> 5, lane = tid & 31;
  const int ln = lane & 15, hi = lane >> 4;
  const int m0 = blockIdx.x * 64, n0 = blockIdx.y * 128;
  const int wr = wave >> 2, wc = wave & 3;

  // Per-thread staging coordinates (A: 2 x b128, B: 4 x b128 per chunk)
  const int ar = tid >> 3, ac = (tid & 7) << 2;

  auto issue_chunk = [&](int p, int k0) {
#pragma unroll
    for (int l = 0; l < 2; ++l) {
      int fid = tid + l * 256;
      int r = fid >> 3, c = (fid & 7) << 2;
      async_copy_b128(A + (size_t)(m0 + r) * K + k0 + c,
                      &As[p][r * GEMM_LDA + c]);
    }
#pragma unroll
    for (int l = 0; l < 4; ++l) {
      int fid = tid + l * 256;
      int r = fid >> 3, c = (fid & 7) << 2;
      async_copy_b128(W + (size_t)(n0 + r) * K + k0 + c,
                      &Bs[p][r * GEMM_LDA + c]);
    }
  };
  (void)ar; (void)ac;

  v8f acc[2][2] = {};
  const int nchunks = K >> 5;  // K / 32
  issue_chunk(0, 0);
  for (int i = 0; i < nchunks; ++i) {
    const int p = i & 1;
    if (i + 1 < nchunks) {
      issue_chunk(p ^ 1, (i + 1) << 5);
      WAIT_ASYNC(6);  // chunk i complete; chunk i+1 still in flight
    } else {
      WAIT_ASYNC(0);
    }
    __syncthreads();  // chunk i visible to all waves
    const float* Ap = As[p];
    const float* Bp = Bs[p];
#pragma unroll
    for (int kk = 0; kk < 8; ++kk) {
      int ks = kk * 4 + 2 * hi;  // lane-half selects K pair {0,1} / {2,3}
      v2f a0 = *(const v2f*)(Ap + (wr * 32 + ln) * GEMM_LDA + ks);
      v2f a1 = *(const v2f*)(Ap + (wr * 32 + 16 + ln) * GEMM_LDA + ks);
      v2f b0 = *(const v2f*)(Bp + (wc * 32 + ln) * GEMM_LDA + ks);
      v2f b1 = *(const v2f*)(Bp + (wc * 32 + 16 + ln) * GEMM_LDA + ks);
      acc[0][0] = wmma_f32(a0, b0, acc[0][0]);
      acc[0][1] = wmma_f32(a0, b1, acc[0][1]);
      acc[1][0] = wmma_f32(a1, b0, acc[1][0]);
      acc[1][1] = wmma_f32(a1, b1, acc[1][1]);
    }
    __syncthreads();  // all waves done reading buffer p before it is refilled
  }
  // Epilogue: C/D layout -> lane<16: M=v, lane>=16: M=v+8; N = lane&15
#pragma unroll
  for (int ci = 0; ci < 2; ++ci) {
    const int ccol = n0 + wc * 32 + ci * 16 + ln;
    const float bb = bias ? bias[ccol] : 0.0f;
#pragma unroll
    for (int ri = 0; ri < 2; ++ri) {
      const int rbase = m0 + wr * 32 + ri * 16 + (hi ? 8 : 0);
#pragma unroll
      for (int v = 0; v < 8; ++v)
        C[(size_t)(rbase + v) * N + ccol] = acc[ri][ci][v] + bb;
    }
  }
}

// ---------------------------------------------------------------------------
// RMSNorm + RoPE (rotary_dim=64 of head_dim=128), in place on qkv.
// grid = (T, 32 q-heads + 8 k-heads), block = 128.
// ---------------------------------------------------------------------------
__global__ __launch_bounds__(128) void norm_rope(
    float* __restrict__ qkv, const int* __restrict__ positions,
    const float* __restrict__ qw, const float* __restrict__ kw) {
  const int t = blockIdx.x, hh = blockIdx.y, tid = threadIdx.x;
  const float* w;
  int base;
  if (hh < 32) { base = hh * 128;               w = qw; }
  else         { base = 4096 + (hh - 32) * 128; w = kw; }
  float* vec = qkv + (size_t)t * 6144 + base;

  __shared__ float red[128];
  __shared__ float xs[128];
  float x = vec[tid];
  red[tid] = x * x;
  __syncthreads();
#pragma unroll
  for (int s = 64; s > 0; s >>= 1) {
    if (tid < s) red[tid] += red[tid + s];
    __syncthreads();
  }
  const float inv = rsqrtf(red[0] * (1.0f / 128.0f) + 1e-5f);
  const float xn = x * inv * w[tid];
  xs[tid] = xn;
  __syncthreads();

  const float pos = (float)positions[t];
  if (tid < 32) {
    // inv_freq[i] = theta^(-i/32)
    float f = pos * __expf(-((float)tid * (1.0f / 32.0f)) * logf(10000.0f));
    float c = cosf(f), s = sinf(f);
    float x1 = xs[tid], x2 = xs[tid + 32];
    vec[tid]      = x1 * c - x2 * s;
    vec[tid + 32] = x2 * c + x1 * s;
  } else if (tid >= 64) {
    vec[tid] = xn;  // pass-through half (normalized only)
  }
}

// ---------------------------------------------------------------------------
// Flash attention, f32 WMMA. grid = (T/32, 32 heads), block = 256 (8 waves).
// Q tile 32x128 (pre-scaled), key tiles 32x128, online softmax.
// V is staged TRANSPOSED (Vst[dim][key]) so PV B-fragments are one b64 load.
// ---------------------------------------------------------------------------
#define LDH 132    // padded head-dim stride (Q/K, row = token)
#define LDS_S 36   // padded S / Vst stride

__global__ __launch_bounds__(256) void attention(
    const float* __restrict__ qkv, float* __restrict__ attn) {
  __shared__ float Qs[32 * LDH];
  __shared__ float Ks[32 * LDH];
  __shared__ float Vst[128 * LDS_S];  // [dim][key]
  __shared__ float Ss[32 * LDS_S];
  __shared__ float m_s[32], l_s[32], r_s[32];

  const int t0 = blockIdx.x * 32;
  const int h = blockIdx.y, kvh = h >> 2;
  const int tid = threadIdx.x, wave = tid >> 5, lane = tid & 31;
  const int ln = lane & 15, hi = lane >> 4;
  const float scale = 0.08838834764831845f;  // 1/sqrt(128)

  const size_t qcol = (size_t)h * 128;
  const size_t kcol = 4096 + (size_t)kvh * 128;
  const size_t vcol = 5120 + (size_t)kvh * 128;

  // Load + pre-scale Q tile (32x128): 4 float4 per thread
#pragma unroll
  for (int i = 0; i < 4; ++i) {
    int fid = tid + i * 256;
    int r = fid >> 5, c = (fid & 31) << 2;
    float4 v = *(const float4*)(qkv + (size_t)(t0 + r) * 6144 + qcol + c);
    v.x *= scale; v.y *= scale; v.z *= scale; v.w *= scale;
    *(float4*)(Qs + r * LDH + c) = v;
  }
  if (tid < 32) { m_s[tid] = -1e30f; l_s[tid] = 0.0f; }

  const int rw = wave >> 2;        // O row-tile (0..1)
  const int cb = (wave & 3) * 32;  // O col base, 2 col-tiles per wave
  v8f o[2] = {};
  __syncthreads();

  const int nkb = blockIdx.x + 1;  // causal: key blocks 0..t0/32
  for (int kb = 0; kb < nkb; ++kb) {
    const int j0 = kb * 32;
    // Stage K (row-major) and V (transposed) tiles
#pragma unroll
    for (int i = 0; i < 4; ++i) {
      int fid = tid + i * 256;
      int r = fid >> 5, c = (fid & 31) << 2;
      *(float4*)(Ks + r * LDH + c) =
          *(const float4*)(qkv + (size_t)(j0 + r) * 6144 + kcol + c);
      float4 v = *(const float4*)(qkv + (size_t)(j0 + r) * 6144 + vcol + c);
      Vst[(c + 0) * LDS_S + r] = v.x;
      Vst[(c + 1) * LDS_S + r] = v.y;
      Vst[(c + 2) * LDS_S + r] = v.z;
      Vst[(c + 3) * LDS_S + r] = v.w;
    }
    __syncthreads();

    // S = Q @ K^T (32x32): tiles on waves 0..3 (wave-uniform branch)
    if (wave < 4) {
      const int si = wave >> 1, sj = wave & 1;
      v8f s = {};
#pragma unroll
      for (int kk = 0; kk < 32; ++kk) {
        int ks = kk * 4 + 2 * hi;
        v2f a = *(const v2f*)(Qs + (si * 16 + ln) * LDH + ks);
        v2f b = *(const v2f*)(Ks + (sj * 16 + ln) * LDH + ks);
        s = wmma_f32(a, b, s);
      }
      const int qrow = si * 16 + (hi ? 8 : 0);
      const int kc = sj * 16 + ln;
      const int kg = j0 + kc;
#pragma unroll
      for (int v = 0; v < 8; ++v) {
        int qg = t0 + qrow + v;
        Ss[(qrow + v) * LDS_S + kc] = (kg <= qg) ? s[v] : -1e30f;
      }
    }
    __syncthreads();

    // Online softmax update: one thread per query row
    if (tid < 32) {
      float mold = m_s[tid];
      float mx = mold;
      float* srow = Ss + tid * LDS_S;
      for (int c = 0; c < 32; ++c) mx = fmaxf(mx, srow[c]);
      float corr = __expf(mold - mx);
      float sum = 0.0f;
      for (int c = 0; c < 32; ++c) {
        float p = __expf(srow[c] - mx);
        srow[c] = p;
        sum += p;
      }
      l_s[tid] = l_s[tid] * corr + sum;
      m_s[tid] = mx;
      r_s[tid] = corr;
    }
    __syncthreads();

    // Rescale accumulators, then O += P @ V
#pragma unroll
    for (int v = 0; v < 8; ++v) {
      float f = r_s[rw * 16 + (hi ? 8 : 0) + v];
      o[0][v] *= f;
      o[1][v] *= f;
    }
#pragma unroll
    for (int ct = 0; ct < 2; ++ct) {
      const int cbase = cb + ct * 16;
#pragma unroll
      for (int kk = 0; kk < 8; ++kk) {
        int ks = kk * 4 + 2 * hi;
        v2f a = *(const v2f*)(Ss + (rw * 16 + ln) * LDS_S + ks);
        v2f b = *(const v2f*)(Vst + (cbase + ln) * LDS_S + ks);
        o[ct] = wmma_f32(a, b, o[ct]);
      }
    }
    __syncthreads();
  }

  // Finalize: divide by row sum, write attn[t, h*128 + col]
#pragma unroll
  for (int v = 0; v < 8; ++v) {
    int r = rw * 16 + (hi ? 8 : 0) + v;
    float inv = 1.0f / l_s[r];
    int tg = t0 + r;
#pragma unroll
    for (int ct = 0; ct < 2; ++ct) {
      attn[(size_t)tg * 4096 + (size_t)h * 128 + cb + ct * 16 + ln] =
          o[ct][v] * inv;
    }
  }
}

// ---------------------------------------------------------------------------
extern "C" void kernel_launch(void* const* d_in, const int* in_sizes, int n_in,
                              void* d_out, int out_size, void* d_ws,
                              size_t ws_size, hipStream_t stream) {
  (void)in_sizes; (void)n_in; (void)out_size; (void)ws_size;
  const int*   positions = (const int*)d_in[0];
  const float* hidden    = (const float*)d_in[1];
  const float* w_qkv     = (const float*)d_in[2];
  const float* b_qkv     = (const float*)d_in[3];
  const float* q_norm_w  = (const float*)d_in[4];
  const float* k_norm_w  = (const float*)d_in[5];
  const float* w_o       = (const float*)d_in[6];
  float* out = (float*)d_out;

  float* qkv  = (float*)d_ws;               // 2048 x 6144
  float* attn = qkv + (size_t)2048 * 6144;  // 2048 x 4096

  // 1) qkv = hidden @ w_qkv^T + b_qkv   (M=2048, N=6144, K=4096)
  gemm_nt<<<dim3(2048 / 64, 6144 / 128), 256, 0, stream>>>(
      hidden, w_qkv, b_qkv, qkv, 2048, 6144, 4096);

  // 2) RMSNorm + RoPE on q (32 heads) and k (8 heads), in place
  norm_rope<<<dim3(2048, 40), 128, 0, stream>>>(qkv, positions, q_norm_w,
                                                k_norm_w);

  // 3) Flash attention -> attn (2048 x 4096)
  attention<<<dim3(2048 / 32, 32), 256, 0, stream>>>(qkv, attn);

  // 4) out = attn @ w_o^T   (M=2048, N=4096, K=4096)
  gemm_nt<<<dim3(2048 / 64, 4096 / 128), 256, 0, stream>>>(
      attn, w_o, nullptr, out, 2048, 4096, 4096);
}